// RGCNEncoder_15023795602048
// MI455X (gfx1250) — compile-verified
//
#include <hip/hip_runtime.h>
#include <hip/hip_bf16.h>

// Problem constants (from the reference)
#define NN 100000   // nodes
#define CC 160      // channels
#define RR 10       // relations
#define BB 5        // blocks
#define CB 32       // per-block channels (== wave32 width)

#define COLTILES (CC / 16)   // 10 column tiles of 16
#define ROWTILES (NN / 16)   // 6250 row tiles of 16
#define KCHUNK   80          // K-rows of root staged in LDS per phase (2 phases)
#define ROOT_WAVES_PER_BLOCK 10
#define ROOT_THREADS (ROOT_WAVES_PER_BLOCK * 32)   // 320; 625 blocks * 10 waves = 6250 exactly

typedef float v2f __attribute__((ext_vector_type(2)));
typedef float v8f __attribute__((ext_vector_type(8)));

// ---------------------------------------------------------------------------
// Count edges per (relation, dst) segment: cnt[r*N + dst] += 1
// ---------------------------------------------------------------------------
__global__ void rgcn_count_kernel(const int* __restrict__ dst,
                                  const int* __restrict__ etype,
                                  float* __restrict__ cnt, int nEdges) {
    int e = blockIdx.x * blockDim.x + threadIdx.x;
    if (e >= nEdges) return;
    int r = etype[e];
    int d = dst[e];
    atomicAdd(&cnt[(size_t)r * NN + d], 1.0f);
}

// cnt -> 1 / max(cnt, 1)   (in place)
__global__ void rgcn_invcnt_kernel(float* __restrict__ cnt, int n) {
    int i = blockIdx.x * blockDim.x + threadIdx.x;
    if (i >= n) return;
    cnt[i] = 1.0f / fmaxf(cnt[i], 1.0f);
}

// ---------------------------------------------------------------------------
// Per-edge message: y[dst] += (W_{r} x[src]) * inv_cnt[r,dst]
// One wave (32 lanes) per edge; Cb == 32 so lane = output channel in block.
// x (64 MB) and y (64 MB) are L2-resident on MI455X (192 MB L2), so the
// gather + global_atomic_add_f32 scatter stay on-chip.
// ---------------------------------------------------------------------------
__global__ void rgcn_edge_scatter_kernel(const float* __restrict__ x,
                                         const float* __restrict__ w,    // [R,B,Cb,Cb]
                                         const float* __restrict__ inv,  // [R*N]
                                         const int* __restrict__ src,
                                         const int* __restrict__ dst,
                                         const int* __restrict__ etype,
                                         float* __restrict__ y, int nEdges) {
    int wave = (blockIdx.x * blockDim.x + threadIdx.x) >> 5;
    int lane = threadIdx.x & 31;
    if (wave >= nEdges) return;

    int s = src[wave];
    int d = dst[wave];
    int r = etype[wave];
    float scale = inv[(size_t)r * NN + d];

    const float* xs = x + (size_t)s * CC;
    const float* wr = w + (size_t)r * (BB * CB * CB);
    float*       yd = y + (size_t)d * CC;

    for (int b = 0; b < BB; ++b) {
        float xv = xs[b * CB + lane];            // this lane's input channel
        const float* wb = wr + b * CB * CB;      // [c][dout], dout = lane
        float acc = 0.0f;
#pragma unroll
        for (int c = 0; c < CB; ++c) {
            float xc = __shfl(xv, c, 32);        // broadcast x[b*32 + c]
            acc = fmaf(xc, wb[c * CB + lane], acc);
        }
        atomicAdd(&yd[b * CB + lane], acc * scale);
    }
}

// ---------------------------------------------------------------------------
// Dense root GEMM epilogue with native fp32 WMMA:
//   y = act( y + x @ root + bias )
// One wave computes a full 16x160 stripe: 10 accumulators, A loaded once per
// K-step and reused across all 10 column tiles (10 v_wmma_f32_16x16x4_f32 per
// K-step). root (100 KB) is staged into LDS in two 80-row chunks and shared
// by all 10 waves of the block, so B reads are conflict-free ds_loads.
//
// fp32 A 16x4 layout (ISA 7.12.2): lanes 0-15 hold M=0..15; VGPR0 = K {0|2},
// VGPR1 = K {1|3} selected by lane[4]. B 4x16 mirrors with N across lanes.
// C/D: VGPR v = row (v + 8*hi), col = lane&15.
// ---------------------------------------------------------------------------
__global__ void __launch_bounds__(ROOT_THREADS)
rgcn_root_wmma_kernel(const float* __restrict__ x,
                      const float* __restrict__ root,  // [C,C]
                      const float* __restrict__ bias,  // [C]
                      float* __restrict__ y,
                      int doRelu) {
    __shared__ float Bs[KCHUNK * CC];                    // 80*160*4 = 51,200 B

    int wave = threadIdx.x >> 5;
    int lane = threadIdx.x & 31;
    int rowTile = blockIdx.x * ROOT_WAVES_PER_BLOCK + wave;   // exact: 625*10 == 6250
    int row0 = rowTile * 16;
    int lo = lane & 15;
    int hi = lane >> 4;                                  // 0 or 1

    v8f acc[COLTILES] = {};
    const float* xrow = x + (size_t)(row0 + lo) * CC;

    for (int kb = 0; kb < CC; kb += KCHUNK) {
        // cooperative stage of root rows [kb, kb+KCHUNK) into LDS (float4)
        __syncthreads();                                 // chunk from prev phase consumed
        const float4* src4 = (const float4*)(root + (size_t)kb * CC);
        float4* dst4 = (float4*)Bs;
        for (int i = threadIdx.x; i < KCHUNK * CC / 4; i += ROOT_THREADS)
            dst4[i] = src4[i];
        __syncthreads();

        for (int k = kb; k < kb + KCHUNK; k += 4) {
            int ka = k + 2 * hi;
            v2f a;
            a.x = xrow[ka];
            a.y = xrow[ka + 1];
            const float* brow0 = Bs + (size_t)(ka - kb) * CC + lo;
#pragma unroll
            for (int t = 0; t < COLTILES; ++t) {
                v2f b;
                b.x = brow0[t * 16];
                b.y = brow0[t * 16 + CC];
                acc[t] = __builtin_amdgcn_wmma_f32_16x16x4_f32(
                    /*neg_a=*/false, a, /*neg_b=*/false, b,
                    /*c_mod=*/(short)0, acc[t], /*reuse_a=*/false, /*reuse_b=*/false);
            }
        }
    }

    // epilogue: y = act(y + acc + bias) for the whole 16x160 stripe
#pragma unroll
    for (int t = 0; t < COLTILES; ++t) {
        int col = t * 16 + lo;
        float bv = bias[col];
#pragma unroll
        for (int v = 0; v < 8; ++v) {
            int m = v + 8 * hi;
            size_t idx = (size_t)(row0 + m) * CC + col;
            float val = y[idx] + acc[t][v] + bv;         // y holds scattered messages
            if (doRelu) val = fmaxf(val, 0.0f);
            y[idx] = val;
        }
    }
}

// ---------------------------------------------------------------------------
// Host-side orchestration (graph-capture safe: kernels + hipMemsetAsync only)
// ---------------------------------------------------------------------------
extern "C" void kernel_launch(void* const* d_in, const int* in_sizes, int n_in,
                              void* d_out, int out_size, void* d_ws, size_t ws_size,
                              hipStream_t stream) {
    const float* node_emb = (const float*)d_in[0];
    const float* w1       = (const float*)d_in[1];
    const float* root1    = (const float*)d_in[2];
    const float* bias1    = (const float*)d_in[3];
    const float* w2       = (const float*)d_in[4];
    const float* root2    = (const float*)d_in[5];
    const float* bias2    = (const float*)d_in[6];
    const int*   eindex   = (const int*)d_in[7];
    const int*   etype    = (const int*)d_in[8];

    const int E = in_sizes[7] / 2;
    const int* src = eindex;         // edge_index[0]
    const int* dst = eindex + E;     // edge_index[1]

    // Workspace: x1 [N*C] floats, then cnt/inv [R*N] floats  (68 MB total)
    float* x1  = (float*)d_ws;
    float* cnt = (float*)((char*)d_ws + (size_t)NN * CC * sizeof(float));
    float* out = (float*)d_out;

    const int THREADS = 256;
    int edgeThreadBlocks  = (E + THREADS - 1) / THREADS;                 // 1 thread/edge
    int edgeWaveBlocks    = ((size_t)E * 32 + THREADS - 1) / THREADS;    // 1 wave/edge
    int segBlocks         = (RR * NN + THREADS - 1) / THREADS;
    int rootBlocks        = ROWTILES / ROOT_WAVES_PER_BLOCK;             // 625 exactly

    // --- shared segment counts -> inverse counts ---
    hipMemsetAsync(cnt, 0, (size_t)RR * NN * sizeof(float), stream);
    rgcn_count_kernel<<<edgeThreadBlocks, THREADS, 0, stream>>>(dst, etype, cnt, E);
    rgcn_invcnt_kernel<<<segBlocks, THREADS, 0, stream>>>(cnt, RR * NN);

    // --- layer 1: x1 = relu( scatter(node_emb) + node_emb @ root1 + bias1 ) ---
    hipMemsetAsync(x1, 0, (size_t)NN * CC * sizeof(float), stream);
    rgcn_edge_scatter_kernel<<<edgeWaveBlocks, THREADS, 0, stream>>>(
        node_emb, w1, cnt, src, dst, etype, x1, E);
    rgcn_root_wmma_kernel<<<rootBlocks, ROOT_THREADS, 0, stream>>>(
        node_emb, root1, bias1, x1, /*doRelu=*/1);

    // --- layer 2: out = scatter(x1) + x1 @ root2 + bias2 ---
    hipMemsetAsync(out, 0, (size_t)NN * CC * sizeof(float), stream);
    rgcn_edge_scatter_kernel<<<edgeWaveBlocks, THREADS, 0, stream>>>(
        x1, w2, cnt, src, dst, etype, out, E);
    rgcn_root_wmma_kernel<<<rootBlocks, ROOT_THREADS, 0, stream>>>(
        x1, root2, bias2, out, /*doRelu=*/0);
}